// GAT_45208825757771
// MI455X (gfx1250) — compile-verified
//
#include <hip/hip_runtime.h>
#include <math.h>

// ---------------- problem constants (from reference) ----------------
#define N_NODES   100000
#define N_EDGES   1600000
#define E_TOT     (N_EDGES + N_NODES)   // self-loops appended
#define N_FEAT    128
#define N_CLASSES 10
#define N_GRAPHS  256

typedef __attribute__((ext_vector_type(2))) float v2f;
typedef __attribute__((ext_vector_type(8))) float v8f;

// ---------------- generic fill ----------------
__global__ void fill_kernel(float* __restrict__ p, long long n, float v) {
    long long t = (long long)blockIdx.x * blockDim.x + threadIdx.x;
    if (t < n) p[t] = v;
}

// ---------------- fp32 WMMA GEMM: C[M,N] = A[M,K] @ B[K,N] ----------------
// Block = 256 threads = 8 waves; each wave computes a 16 x N strip.
// B staged in LDS in k-paired float2 layout: Bs[k/2][col] = (B[k][col], B[k+1][col])
// so every B fragment is one aligned ds_load_b64 straight into an even VGPR pair.
// Fragment layouts per CDNA5 ISA 7.12.2:
//   A 16x4 : lanes 0-15 -> M=lane,  v0=K0,v1=K1 ; lanes 16-31 -> M=lane-16, v0=K2,v1=K3
//   B 4x16 : lanes 0-15 -> N=lane,  v0=K0,v1=K1 ; lanes 16-31 -> N=lane-16, v0=K2,v1=K3
//   C 16x16: VGPR v -> M=v (lanes 0-15) / M=v+8 (lanes 16-31), N = lane&15
template<int K, int N>
__global__ __launch_bounds__(256) void gemm_wmma_f32(const float* __restrict__ A,
                                                     const float* __restrict__ B,
                                                     float* __restrict__ C, int M) {
    constexpr int NT = N / 16;
    __shared__ v2f Bs[(K / 2) * N];
    for (int i = threadIdx.x; i < (K / 2) * N; i += 256) {
        const int kp = i / N, col = i % N;
        v2f t;
        t.x = B[(2 * kp) * N + col];
        t.y = B[(2 * kp + 1) * N + col];
        Bs[i] = t;
    }
    __syncthreads();

    const int wave = threadIdx.x >> 5;
    const int lane = threadIdx.x & 31;
    const int rowBase = (blockIdx.x * 8 + wave) * 16;
    int arow = rowBase + (lane & 15);
    if (arow >= M) arow = M - 1;            // clamp: keep EXEC all-ones for WMMA
    const int koff = (lane >> 4) << 1;      // 0 for lanes 0-15, 2 for lanes 16-31
    const int col  = lane & 15;

    const float* Arow = A + (size_t)arow * K;
    __builtin_prefetch(Arow, 0, 3);         // global_prefetch_b8: pull this wave's A strip

    v8f acc[NT] = {};

#pragma unroll 4
    for (int kt = 0; kt < K / 4; ++kt) {
        const int k = kt * 4 + koff;
        const v2f a = *(const v2f*)(Arow + k);            // one global_load_b64
#pragma unroll
        for (int nt = 0; nt < NT; ++nt) {
            const v2f b = Bs[(k >> 1) * N + nt * 16 + col]; // one ds_load_b64
            acc[nt] = __builtin_amdgcn_wmma_f32_16x16x4_f32(
                /*neg_a=*/false, a, /*neg_b=*/false, b,
                /*c_mod=*/(short)0, acc[nt], /*reuse_a=*/false, /*reuse_b=*/false);
        }
    }

    const int srow0 = rowBase + ((lane >> 4) << 3);
#pragma unroll
    for (int v = 0; v < 8; ++v) {
        const int r = srow0 + v;
        if (r < M) {
#pragma unroll
            for (int nt = 0; nt < NT; ++nt)
                C[(size_t)r * N + nt * 16 + col] = acc[nt][v];
        }
    }
}

// ---------------- per-node attention coefficients ----------------
// al_s[n,h] = sum_c h[n,h*C+c]*a_src[h,c] ; al_d likewise
template<int H, int C>
__global__ void attn_coef_kernel(const float* __restrict__ h,
                                 const float* __restrict__ a_src,
                                 const float* __restrict__ a_dst,
                                 float* __restrict__ al_s,
                                 float* __restrict__ al_d, int n) {
    int t = blockIdx.x * blockDim.x + threadIdx.x;
    if (t >= n * H) return;
    const int node = t / H, hh = t % H;
    const float* row = h + (size_t)node * (H * C) + hh * C;
    float ss = 0.f, dd = 0.f;
#pragma unroll
    for (int c = 0; c < C; ++c) {
        float v = row[c];
        ss += v * a_src[hh * C + c];
        dd += v * a_dst[hh * C + c];
    }
    al_s[t] = ss;
    al_d[t] = dd;
}

// ---------------- edge helpers ----------------
__device__ __forceinline__ void edge_nodes(const long long* __restrict__ ei,
                                           long long e, int& s, int& d) {
    if (e < N_EDGES) { s = (int)ei[e]; d = (int)ei[N_EDGES + e]; }
    else             { s = d = (int)(e - N_EDGES); }   // self-loop
}

__device__ __forceinline__ float leaky(float v) { return v > 0.f ? v : 0.2f * v; }

// pass 1: m[dst,h] = max over incoming edges of leaky(al_s[src]+al_d[dst])
template<int H>
__global__ void attn_max_kernel(const long long* __restrict__ ei,
                                const float* __restrict__ als,
                                const float* __restrict__ ald,
                                float* __restrict__ m) {
    long long t = (long long)blockIdx.x * blockDim.x + threadIdx.x;
    if (t >= (long long)E_TOT * H) return;
    const long long e = t / H; const int h = (int)(t % H);
    int s, d; edge_nodes(ei, e, s, d);
    const float v = leaky(als[(size_t)s * H + h] + ald[(size_t)d * H + h]);
    atomicMax(&m[(size_t)d * H + h], v);
}

// pass 2: ssum[dst,h] = sum exp(e - m[dst,h])
template<int H>
__global__ void attn_sum_kernel(const long long* __restrict__ ei,
                                const float* __restrict__ als,
                                const float* __restrict__ ald,
                                const float* __restrict__ m,
                                float* __restrict__ ssum) {
    long long t = (long long)blockIdx.x * blockDim.x + threadIdx.x;
    if (t >= (long long)E_TOT * H) return;
    const long long e = t / H; const int h = (int)(t % H);
    int s, d; edge_nodes(ei, e, s, d);
    const float v = leaky(als[(size_t)s * H + h] + ald[(size_t)d * H + h]);
    float mm = m[(size_t)d * H + h];
    if (!isfinite(mm)) mm = 0.f;
    atomicAdd(&ssum[(size_t)d * H + h], expf(v - mm));
}

// pass 3: out[dst,f] += h[src,f] * alpha(e,h), alpha recomputed (saves a 54MB
// per-edge alpha array round-trip). One WAVE per edge: the edge id is wave-
// uniform (edge_index loads scalarize), lanes stride the F features, h[src]
// row reads stay coalesced, scatter atomics stay L2-resident.
template<int H, int C>   // F = H*C features
__global__ __launch_bounds__(256) void aggregate_kernel(const long long* __restrict__ ei,
                                 const float* __restrict__ als,
                                 const float* __restrict__ ald,
                                 const float* __restrict__ m,
                                 const float* __restrict__ ssum,
                                 const float* __restrict__ hfeat,
                                 float* __restrict__ out) {
    constexpr int F = H * C;
    const int lane = threadIdx.x & 31;
    const long long e = (long long)blockIdx.x * 8 + (threadIdx.x >> 5);
    if (e >= E_TOT) return;
    int s, d; edge_nodes(ei, e, s, d);
    const float* __restrict__ hrow = hfeat + (size_t)s * F;
    float* __restrict__ orow = out + (size_t)d * F;
#pragma unroll
    for (int f = lane; f < F; f += 32) {
        const int h = f / C;                               // 0 when H==1
        const float v = leaky(als[(size_t)s * H + h] + ald[(size_t)d * H + h]);
        float mm = m[(size_t)d * H + h];
        if (!isfinite(mm)) mm = 0.f;
        const float coef = expf(v - mm) / (ssum[(size_t)d * H + h] + 1e-16f);
        atomicAdd(&orow[f], hrow[f] * coef);
    }
}

// ---------------- bias (+ optional ReLU), in place ----------------
__global__ void bias_act_kernel(float* __restrict__ x, const float* __restrict__ b,
                                long long n, int F, int relu) {
    long long t = (long long)blockIdx.x * blockDim.x + threadIdx.x;
    if (t >= n) return;
    float v = x[t] + b[t % F];
    if (relu) v = v > 0.f ? v : 0.f;
    x[t] = v;
}

// ---------------- mean pooling over batch segments ----------------
__global__ void pool_kernel(const float* __restrict__ x2,
                            const long long* __restrict__ batch,
                            float* __restrict__ pooled, float* __restrict__ cnt) {
    long long t = (long long)blockIdx.x * blockDim.x + threadIdx.x;
    if (t >= (long long)N_NODES * N_FEAT) return;
    const int node = (int)(t >> 7), f = (int)(t & 127);
    const int g = (int)batch[node];
    atomicAdd(&pooled[(size_t)g * N_FEAT + f], x2[t]);
    if (f == 0) atomicAdd(&cnt[g], 1.0f);
}

// ---------------- classifier head + log_softmax ----------------
__global__ void head_kernel(const float* __restrict__ pooled, const float* __restrict__ cnt,
                            const float* __restrict__ fcw, const float* __restrict__ fcb,
                            float* __restrict__ out) {
    int g = blockIdx.x * blockDim.x + threadIdx.x;
    if (g >= N_GRAPHS) return;
    float c = cnt[g]; c = c > 1.f ? c : 1.f;
    float logits[N_CLASSES];
#pragma unroll
    for (int j = 0; j < N_CLASSES; ++j) logits[j] = fcb[j];
    for (int k = 0; k < N_FEAT; ++k) {
        const float p = pooled[(size_t)g * N_FEAT + k] / c;
#pragma unroll
        for (int j = 0; j < N_CLASSES; ++j) logits[j] += p * fcw[k * N_CLASSES + j];
    }
    float mx = logits[0];
#pragma unroll
    for (int j = 1; j < N_CLASSES; ++j) mx = fmaxf(mx, logits[j]);
    float se = 0.f;
#pragma unroll
    for (int j = 0; j < N_CLASSES; ++j) se += expf(logits[j] - mx);
    const float lse = mx + logf(se);
#pragma unroll
    for (int j = 0; j < N_CLASSES; ++j) out[(size_t)g * N_CLASSES + j] = logits[j] - lse;
}

// ---------------- launcher ----------------
static inline unsigned blk(long long n, int b) { return (unsigned)((n + b - 1) / b); }

extern "C" void kernel_launch(void* const* d_in, const int* in_sizes, int n_in,
                              void* d_out, int out_size, void* d_ws, size_t ws_size,
                              hipStream_t stream) {
    const float*     x      = (const float*)d_in[0];
    const long long* ei     = (const long long*)d_in[1];
    const long long* batch  = (const long long*)d_in[2];
    const float*     W1     = (const float*)d_in[3];
    const float*     a_src1 = (const float*)d_in[4];
    const float*     a_dst1 = (const float*)d_in[5];
    const float*     b1     = (const float*)d_in[6];
    const float*     W2     = (const float*)d_in[7];
    const float*     a_src2 = (const float*)d_in[8];
    const float*     a_dst2 = (const float*)d_in[9];
    const float*     b2     = (const float*)d_in[10];
    const float*     fc_w   = (const float*)d_in[11];
    const float*     fc_b   = (const float*)d_in[12];
    float*           out    = (float*)d_out;

    // workspace carve-up (all fp32)
    float* ws = (float*)d_ws;
    size_t off = 0;
    float* h1   = ws + off; off += (size_t)N_NODES * 64;
    float* als1 = ws + off; off += (size_t)N_NODES * 8;
    float* ald1 = ws + off; off += (size_t)N_NODES * 8;
    float* m1   = ws + off; off += (size_t)N_NODES * 8;
    float* s1   = ws + off; off += (size_t)N_NODES * 8;
    float* agg1 = ws + off; off += (size_t)N_NODES * 64;   // becomes x1 in place
    float* h2   = ws + off; off += (size_t)N_NODES * 128;
    float* als2 = ws + off; off += (size_t)N_NODES;
    float* ald2 = ws + off; off += (size_t)N_NODES;
    float* m2   = ws + off; off += (size_t)N_NODES;
    float* s2   = ws + off; off += (size_t)N_NODES;
    float* agg2 = ws + off; off += (size_t)N_NODES * 128;  // becomes x2 in place
    float* pooled = ws + off; off += (size_t)N_GRAPHS * N_FEAT;
    float* cnt    = ws + off; off += (size_t)N_GRAPHS;

    const int B = 256;
    const unsigned edgeWaveBlocks = blk((long long)E_TOT, 8);  // 8 edges (waves) per block

    // ---- layer 1 ----
    fill_kernel<<<blk((long long)N_NODES * 8, B), B, 0, stream>>>(m1, (long long)N_NODES * 8, -INFINITY);
    fill_kernel<<<blk((long long)N_NODES * 8, B), B, 0, stream>>>(s1, (long long)N_NODES * 8, 0.f);
    fill_kernel<<<blk((long long)N_NODES * 64, B), B, 0, stream>>>(agg1, (long long)N_NODES * 64, 0.f);

    gemm_wmma_f32<128, 64><<<blk(N_NODES, 128), 256, 0, stream>>>(x, W1, h1, N_NODES);
    attn_coef_kernel<8, 8><<<blk((long long)N_NODES * 8, B), B, 0, stream>>>(h1, a_src1, a_dst1, als1, ald1, N_NODES);
    attn_max_kernel<8><<<blk((long long)E_TOT * 8, B), B, 0, stream>>>(ei, als1, ald1, m1);
    attn_sum_kernel<8><<<blk((long long)E_TOT * 8, B), B, 0, stream>>>(ei, als1, ald1, m1, s1);
    aggregate_kernel<8, 8><<<edgeWaveBlocks, B, 0, stream>>>(ei, als1, ald1, m1, s1, h1, agg1);
    bias_act_kernel<<<blk((long long)N_NODES * 64, B), B, 0, stream>>>(agg1, b1, (long long)N_NODES * 64, 64, 1);

    // ---- layer 2 ----
    fill_kernel<<<blk(N_NODES, B), B, 0, stream>>>(m2, N_NODES, -INFINITY);
    fill_kernel<<<blk(N_NODES, B), B, 0, stream>>>(s2, N_NODES, 0.f);
    fill_kernel<<<blk((long long)N_NODES * 128, B), B, 0, stream>>>(agg2, (long long)N_NODES * 128, 0.f);
    fill_kernel<<<blk((long long)N_GRAPHS * N_FEAT, B), B, 0, stream>>>(pooled, (long long)N_GRAPHS * N_FEAT, 0.f);
    fill_kernel<<<blk(N_GRAPHS, B), B, 0, stream>>>(cnt, N_GRAPHS, 0.f);

    gemm_wmma_f32<64, 128><<<blk(N_NODES, 128), 256, 0, stream>>>(agg1, W2, h2, N_NODES);
    attn_coef_kernel<1, 128><<<blk(N_NODES, B), B, 0, stream>>>(h2, a_src2, a_dst2, als2, ald2, N_NODES);
    attn_max_kernel<1><<<blk((long long)E_TOT, B), B, 0, stream>>>(ei, als2, ald2, m2);
    attn_sum_kernel<1><<<blk((long long)E_TOT, B), B, 0, stream>>>(ei, als2, ald2, m2, s2);
    aggregate_kernel<1, 128><<<edgeWaveBlocks, B, 0, stream>>>(ei, als2, ald2, m2, s2, h2, agg2);
    bias_act_kernel<<<blk((long long)N_NODES * 128, B), B, 0, stream>>>(agg2, b2, (long long)N_NODES * 128, 128, 0);

    // ---- pooling + head ----
    pool_kernel<<<blk((long long)N_NODES * N_FEAT, B), B, 0, stream>>>(agg2, batch, pooled, cnt);
    head_kernel<<<1, N_GRAPHS, 0, stream>>>(pooled, cnt, fc_w, fc_b, out);
}